// GCN_28802050687441
// MI455X (gfx1250) — compile-verified
//
#include <hip/hip_runtime.h>
#include <math.h>

typedef __attribute__((ext_vector_type(2))) float v2f;
typedef __attribute__((ext_vector_type(8))) float v8f;

#define IN_F    128
#define HID_F   128
#define OUT_F   40
#define OUT_PAD 48   // layer-2 GEMM output stored padded to 48 cols (zero-filled)

// ---------------- elementwise helpers ----------------

__global__ void k_fill(float* __restrict__ p, float v, int n) {
  int i = blockIdx.x * blockDim.x + threadIdx.x;
  if (i < n) p[i] = v;
}

__global__ void k_init_out(float* __restrict__ out, const float* __restrict__ b2, int n) {
  int i = blockIdx.x * blockDim.x + threadIdx.x;
  if (i < n) out[i] = b2[i % OUT_F];
}

__global__ void k_deg_scatter(const int* __restrict__ dst, float* __restrict__ deg, int E) {
  int e = blockIdx.x * blockDim.x + threadIdx.x;
  if (e < E) atomicAdd(&deg[dst[e]], 1.0f);
}

__global__ void k_rsqrt(float* __restrict__ deg, int n) {
  int i = blockIdx.x * blockDim.x + threadIdx.x;
  if (i < n) {
    float d = deg[i];
    deg[i] = (d > 0.0f) ? rsqrtf(d) : 0.0f;
  }
}

__global__ void k_relu_bias(float* __restrict__ a, const float* __restrict__ b1, int n) {
  int i = blockIdx.x * blockDim.x + threadIdx.x;
  if (i < n) {
    float v = a[i] + b1[i & (HID_F - 1)];
    a[i] = v > 0.0f ? v : 0.0f;
  }
}

// ---------------- fp32 WMMA GEMM: C[M x NPAD] = A[M x 128] * B[128 x NOUT] ----------------
// Block = 256 threads (8 waves) computes a 64-row strip of C.
// A strip staged in LDS (row stride 132 floats -> 16 lanes on distinct banks).
// FULL (NOUT%16==0): B read straight from global (L2-hot weights), no guards.
// Tail (NOUT=40):    B staged zero-padded + transposed into LDS once -> branch-free K loop,
//                    and C is stored with NPAD stride so stores need no column guard
//                    (padded columns receive exact zeros).
// M-tail handled by a wave-uniform branch: only the last block takes the guarded path.
template <int NOUT>
__global__ __launch_bounds__(256) void k_gemm_wmma(const float* __restrict__ A,
                                                   const float* __restrict__ B,
                                                   float* __restrict__ C, int M) {
  constexpr int  NPAD = (NOUT + 15) & ~15;
  constexpr bool FULL = (NOUT % 16 == 0);
  constexpr int  ROWS = 64;                 // M rows per block
  constexpr int  MT   = ROWS / 16;          // 4 m-tiles
  constexpr int  NT   = NPAD / 16;          // n-tiles
  constexpr int  TILES = MT * NT;

  __shared__ float As[ROWS][132];
  __shared__ float Bst[FULL ? 1 : NPAD][FULL ? 1 : 132];

  const int tid    = threadIdx.x;
  const int m_base = blockIdx.x * ROWS;

  // stage A strip (clamp row; stores are guarded on the tail path)
  for (int idx = tid; idx < ROWS * 128; idx += 256) {
    int r = idx >> 7, c = idx & 127;
    int row = m_base + r;
    if (row >= M) row = M - 1;
    As[r][c] = A[(size_t)row * 128 + c];
  }
  // stage B transposed + zero-padded (tail case only)
  if constexpr (!FULL) {
    for (int idx = tid; idx < NPAD * 128; idx += 256) {
      int k = idx / NPAD, n = idx % NPAD;   // consecutive threads -> consecutive n (coalesced)
      Bst[n][k] = (n < NOUT) ? B[(size_t)k * NOUT + n] : 0.0f;
    }
  }
  __syncthreads();

  const int wave = tid >> 5;
  const int lane = tid & 31;
  const int half = lane >> 4;               // 0: K {k0,k0+1}, 1: K {k0+2,k0+3}
  const int r    = lane & 15;
  const bool mfull = (m_base + ROWS <= M);  // wave-uniform

  for (int t = wave; t < TILES; t += 8) {
    const int mt = t % MT;
    const int nt = t / MT;
    const int n  = nt * 16 + r;
    const float* ap = &As[mt * 16 + r][half * 2];

    v8f acc = {};
    if constexpr (FULL) {
      const float* bp = B + (size_t)(half * 2) * NOUT + n;
      #pragma unroll 4
      for (int kk = 0; kk < 128; kk += 4) {
        v2f a = *(const v2f*)(ap + kk);
        v2f b;
        b.x = bp[(size_t)kk * NOUT];
        b.y = bp[(size_t)kk * NOUT + NOUT];
        acc = __builtin_amdgcn_wmma_f32_16x16x4_f32(false, a, false, b, (short)0, acc,
                                                    false, false);
      }
    } else {
      const float* bp = &Bst[n][half * 2];
      #pragma unroll 4
      for (int kk = 0; kk < 128; kk += 4) {
        v2f a = *(const v2f*)(ap + kk);
        v2f b = *(const v2f*)(bp + kk);
        acc = __builtin_amdgcn_wmma_f32_16x16x4_f32(false, a, false, b, (short)0, acc,
                                                    false, false);
      }
    }

    const int rowb = m_base + mt * 16 + half * 8;
    float* cp = C + (size_t)rowb * NPAD + n;
    if (mfull) {                             // fast path: 781/782 blocks, no guards
      #pragma unroll
      for (int j = 0; j < 8; ++j) cp[(size_t)j * NPAD] = acc[j];
    } else {                                 // M tail: guarded
      #pragma unroll
      for (int j = 0; j < 8; ++j)
        if (rowb + j < M) cp[(size_t)j * NPAD] = acc[j];
    }
  }
}

// ---------------- edge scatter, 128 feats: one wave per edge, float4 per lane ----------------

__global__ void k_scatter128(const int* __restrict__ src, const int* __restrict__ dst,
                             const float* __restrict__ dis, const float* __restrict__ h,
                             float* __restrict__ agg, int E, int N) {
  int gid  = blockIdx.x * blockDim.x + threadIdx.x;
  int w    = gid >> 5;
  int lane = gid & 31;
  if (w >= E + N) return;
  int s, d;
  if (w < E) { s = src[w]; d = dst[w]; }
  else       { s = w - E;  d = s; }           // self-loop
  float norm = dis[s] * dis[d];
  const float4* row = (const float4*)(h + (size_t)s * 128);
  float4 v = row[lane];                        // global_load_b128 from L2-resident h
  float* o = agg + (size_t)d * 128 + lane * 4;
  atomicAdd(o + 0, v.x * norm);
  atomicAdd(o + 1, v.y * norm);
  atomicAdd(o + 2, v.z * norm);
  atomicAdd(o + 3, v.w * norm);
}

// ---------------- edge scatter, 40 feats (h stored with OUT_PAD stride) ----------------

__global__ void k_scatter40(const int* __restrict__ src, const int* __restrict__ dst,
                            const float* __restrict__ dis, const float* __restrict__ h,
                            float* __restrict__ out, int E, int N) {
  int gid  = blockIdx.x * blockDim.x + threadIdx.x;
  int w    = gid >> 5;
  int lane = gid & 31;
  if (w >= E + N) return;
  int s, d;
  if (w < E) { s = src[w]; d = dst[w]; }
  else       { s = w - E;  d = s; }
  float norm = dis[s] * dis[d];
  const float* hr = h + (size_t)s * OUT_PAD;
  float* o = out + (size_t)d * OUT_F;
  for (int f = lane; f < OUT_F; f += 32)
    atomicAdd(&o[f], hr[f] * norm);
}

// ---------------- launcher ----------------

extern "C" void kernel_launch(void* const* d_in, const int* in_sizes, int n_in,
                              void* d_out, int out_size, void* d_ws, size_t ws_size,
                              hipStream_t stream) {
  const float* x  = (const float*)d_in[0];
  const int*   ei = (const int*)d_in[1];
  const float* W1 = (const float*)d_in[2];
  const float* b1 = (const float*)d_in[3];
  const float* W2 = (const float*)d_in[4];
  const float* b2 = (const float*)d_in[5];
  float* out = (float*)d_out;

  const int N = in_sizes[0] / IN_F;
  const int E = in_sizes[1] / 2;
  const int* src = ei;
  const int* dst = ei + E;

  // workspace layout (floats): deg/dis | h0 | agg1(->h) | h1 (padded to OUT_PAD)
  size_t nAl = ((size_t)N + 63) & ~(size_t)63;
  float* dis = (float*)d_ws;
  float* h0  = dis + nAl;
  float* agg = h0 + (size_t)N * HID_F;
  float* h1  = agg + (size_t)N * HID_F;

  const int T = 256;
  int nFeat1 = N * HID_F;
  int nOut   = N * OUT_F;

  // degrees (self-loop => start at 1.0), agg1 zero, out preloaded with bias b2
  k_fill<<<(N + T - 1) / T, T, 0, stream>>>(dis, 1.0f, N);
  k_fill<<<(nFeat1 + T - 1) / T, T, 0, stream>>>(agg, 0.0f, nFeat1);
  k_init_out<<<(nOut + T - 1) / T, T, 0, stream>>>(out, b2, nOut);
  k_deg_scatter<<<(E + T - 1) / T, T, 0, stream>>>(dst, dis, E);
  k_rsqrt<<<(N + T - 1) / T, T, 0, stream>>>(dis, N);

  const int gemmBlocks = (N + 63) / 64;

  // layer 1: h0 = x @ W1 (fp32 WMMA), scatter into agg, fused bias+ReLU in place
  k_gemm_wmma<HID_F><<<gemmBlocks, 256, 0, stream>>>(x, W1, h0, N);
  {
    long long waves = (long long)E + N;
    long long thr   = waves * 32;
    int blocks = (int)((thr + T - 1) / T);
    k_scatter128<<<blocks, T, 0, stream>>>(src, dst, dis, h0, agg, E, N);
  }
  k_relu_bias<<<(nFeat1 + T - 1) / T, T, 0, stream>>>(agg, b1, nFeat1);

  // layer 2: h1 = h @ W2 (N=40 padded to 48 in LDS and in the output buffer),
  // scatter into out (bias pre-set)
  k_gemm_wmma<OUT_F><<<gemmBlocks, 256, 0, stream>>>(agg, W2, h1, N);
  {
    long long waves = (long long)E + N;
    long long thr   = waves * 32;
    int blocks = (int)((thr + T - 1) / T);
    k_scatter40<<<blocks, T, 0, stream>>>(src, dst, dis, h1, out, E, N);
  }
}